// EfficientVQGAN_54949811585632
// MI455X (gfx1250) — compile-verified
//
#include <hip/hip_runtime.h>
#include <hip/hip_bf16.h>

typedef __attribute__((ext_vector_type(2))) float v2f;
typedef __attribute__((ext_vector_type(8))) float v8f;

#define DDIM   128
#define KCODES 8192
#define NROWS  16384
#define DECAYF 0.99f
#define OMDEC  0.01f
#define EPSF   1e-5f
#define BETAF  0.25f

// ---- d_out layout (float element offsets), concatenated reference outputs ----
#define OUT_ZQ     0                          // [16,128,32,32] = 2097152
#define OUT_IDX    2097152                    // [16384] int32
#define OUT_LOSS   (OUT_IDX + NROWS)          // scalar
#define OUT_NEWEMB (OUT_LOSS + 1)             // [8192,128]
#define OUT_NEWCS  (OUT_NEWEMB + KCODES*DDIM) // [8192]
#define OUT_NEWW   (OUT_NEWCS + KCODES)       // [8192,128]

// ---- workspace layout (float element offsets) ----
#define WS_CLUSTER 0
#define WS_EMBSUM  (WS_CLUSTER + KCODES)
#define WS_LOSS    (WS_EMBSUM + KCODES*DDIM)
#define WS_N       (WS_LOSS + 1)
#define WS_W2      (WS_N + 1)
#define WS_ZERO_CT (KCODES + KCODES*DDIM + 2)

// -------------------------------------------------------------------------
// Kernel 0: zero the accumulator scratch (must run every launch).
__global__ __launch_bounds__(256) void vq_zero(float* __restrict__ ws, unsigned long long n) {
    unsigned long long i = (unsigned long long)blockIdx.x * 256ull + threadIdx.x;
    unsigned long long stride = (unsigned long long)gridDim.x * 256ull;
    for (; i < n; i += stride) ws[i] = 0.0f;
}

// -------------------------------------------------------------------------
// Kernel 1: w2[k] = sum_d emb[k][d]^2.  One wave32 per code row.
__global__ __launch_bounds__(256) void vq_w2(const float* __restrict__ emb,
                                             float* __restrict__ w2) {
    const int wave = threadIdx.x >> 5;
    const int lane = threadIdx.x & 31;
    const int k = blockIdx.x * 8 + wave;           // grid 1024 * 8 waves = 8192
    const float4 v = *((const float4*)(emb + (size_t)k * DDIM) + lane);
    float s = v.x * v.x + v.y * v.y + v.z * v.z + v.w * v.w;
#pragma unroll
    for (int m = 16; m >= 1; m >>= 1) s += __shfl_xor(s, m, 32);
    if (lane == 0) w2[k] = s;
}

// -------------------------------------------------------------------------
// Kernel 2: fp32 WMMA distance GEMM + fused argmin.
// Block = 256 threads (8 waves), handles 16 rows of `flat`; each wave scans
// a disjoint 1024-code range with V_WMMA_F32_16X16X4_F32.
// B fragments are software-pipelined in two half-tiles (16 k-steps each) so
// global_load_b64s overlap the serial WMMA accumulation chain instead of
// forcing s_wait_loadcnt 0 before every WMMA.
__global__ __launch_bounds__(256) void vq_dist_argmin(const float* __restrict__ z,
                                                      const float* __restrict__ emb,
                                                      const float* __restrict__ w2,
                                                      int* __restrict__ out_idx) {
    __shared__ float tileA[16][130];   // padded: avoids 64-bank conflicts on ds_load_b64
    __shared__ float x2s[16];
    __shared__ float redD[8][16];
    __shared__ int   redI[8][16];

    const int t  = threadIdx.x;
    const int i0 = blockIdx.x * 16;    // 1024 blocks * 16 rows = 16384
    // flat[i][d] = z[(i>>10)*131072 + d*1024 + (i&1023)]  (NCHW -> NHWC transpose)
    const long zbase = (long)(i0 >> 10) * (DDIM * 1024) + (i0 & 1023);

    // Stage A tile [16 rows][128 d] with coalesced 64B runs.
#pragma unroll
    for (int rep = 0; rep < 8; ++rep) {
        const int e = t + rep * 256;
        const int d = e >> 4, r = e & 15;
        tileA[r][d] = z[zbase + (long)d * 1024 + r];
    }
    __syncthreads();
    if (t < 16) {
        float s = 0.0f;
        for (int d = 0; d < DDIM; ++d) { const float v = tileA[t][d]; s += v * v; }
        x2s[t] = s;
    }
    __syncthreads();

    const int wave = t >> 5;
    const int lane = t & 31;
    const int half = lane >> 4;        // 0: lanes 0-15, 1: lanes 16-31
    const int lr   = lane & 15;
    const int koff = half << 1;        // A/B frag K offset: {0,1} vs {2,3}

    // A fragments for all 32 K-steps, per 16x4 fp32 WMMA layout.
    v2f afrag[32];
#pragma unroll
    for (int kk = 0; kk < 32; ++kk)
        afrag[kk] = *(const v2f*)&tileA[lr][4 * kk + koff];

    float x2r[8];
#pragma unroll
    for (int j = 0; j < 8; ++j) x2r[j] = x2s[j + 8 * half];

    float best[8];
    int   bidx[8];
#pragma unroll
    for (int j = 0; j < 8; ++j) { best[j] = 3.4e38f; bidx[j] = 0; }

    const int cbase0 = wave * 1024;    // 8 waves split the 8192 codes

    // Double-buffered B fragments (distinct registers -> pipelined loads).
    v2f bfrag0[16];
    v2f bfrag1[16];
    {
        const float* eb0 = emb + (size_t)(cbase0 + lr) * DDIM + koff;
#pragma unroll
        for (int kk = 0; kk < 16; ++kk)
            bfrag0[kk] = *(const v2f*)(eb0 + 4 * kk);
    }

    for (int tt = 0; tt < 64; ++tt) {
        const int nb = cbase0 + tt * 16;
        const float* eb = emb + (size_t)(nb + lr) * DDIM + koff;
        const float w2n = w2[nb + lr];
        const int   n   = nb + lr;

        // Issue loads for second half of this tile (k-steps 16..31).
#pragma unroll
        for (int kk = 0; kk < 16; ++kk)
            bfrag1[kk] = *(const v2f*)(eb + 64 + 4 * kk);

        // Compute first half while those loads are in flight.
        v8f acc = {0.f, 0.f, 0.f, 0.f, 0.f, 0.f, 0.f, 0.f};
#pragma unroll
        for (int kk = 0; kk < 16; ++kk)
            acc = __builtin_amdgcn_wmma_f32_16x16x4_f32(
                false, afrag[kk], false, bfrag0[kk], (short)0, acc, false, false);

        // Issue loads for first half of the next tile.
        if (tt < 63) {
            const float* ebn = eb + 16 * DDIM;
#pragma unroll
            for (int kk = 0; kk < 16; ++kk)
                bfrag0[kk] = *(const v2f*)(ebn + 4 * kk);
        }

        // Compute second half.
#pragma unroll
        for (int kk = 0; kk < 16; ++kk)
            acc = __builtin_amdgcn_wmma_f32_16x16x4_f32(
                false, afrag[16 + kk], false, bfrag1[kk], (short)0, acc, false, false);

        // Fused distance + running argmin (C layout: VGPR j -> row j/j+8).
#pragma unroll
        for (int j = 0; j < 8; ++j) {
            const float dist = x2r[j] + w2n - 2.0f * acc[j];
            if (dist < best[j]) { best[j] = dist; bidx[j] = n; }
        }
    }

    // Butterfly argmin across the 16 lanes holding the same rows
    // (lanes 0-15: rows 0-7, lanes 16-31: rows 8-15); tie -> lower index.
#pragma unroll
    for (int m = 8; m >= 1; m >>= 1) {
#pragma unroll
        for (int j = 0; j < 8; ++j) {
            const float od = __shfl_xor(best[j], m, 32);
            const int   oi = __shfl_xor(bidx[j], m, 32);
            if (od < best[j] || (od == best[j] && oi < bidx[j])) { best[j] = od; bidx[j] = oi; }
        }
    }
    if (lr == 0) {
#pragma unroll
        for (int j = 0; j < 8; ++j) {
            redD[wave][j + 8 * half] = best[j];
            redI[wave][j + 8 * half] = bidx[j];
        }
    }
    __syncthreads();

    if (t < 16) {   // cross-wave reduce; waves are in ascending code order
        float bd = redD[0][t];
        int   bi = redI[0][t];
#pragma unroll
        for (int w = 1; w < 8; ++w) {
            const float od = redD[w][t];
            const int   oi = redI[w][t];
            if (od < bd || (od == bd && oi < bi)) { bd = od; bi = oi; }
        }
        out_idx[i0 + t] = bi;
    }
}

// -------------------------------------------------------------------------
// Kernel 3: gather z_q (write NCHW), commitment-loss reduction, segment sums.
__global__ __launch_bounds__(256) void vq_gather_scatter(const float* __restrict__ z,
                                                         const float* __restrict__ emb,
                                                         const int* __restrict__ idx,
                                                         float* __restrict__ zq_out,
                                                         float* __restrict__ embed_sum,
                                                         float* __restrict__ cluster,
                                                         float* __restrict__ loss_acc) {
    __shared__ int   sIdx[16];
    __shared__ float sred[256];
    const int t  = threadIdx.x;
    const int i0 = blockIdx.x * 16;
    const long base = (long)(i0 >> 10) * (DDIM * 1024) + (i0 & 1023);

    if (t < 16) {
        const int c = idx[i0 + t];
        sIdx[t] = c;
        atomicAdd(&cluster[c], 1.0f);
    }
    __syncthreads();

    float lsum = 0.0f;
#pragma unroll
    for (int rep = 0; rep < 8; ++rep) {
        const int e = t + rep * 256;
        const int d = e >> 4, r = e & 15;
        const int c = sIdx[r];
        const float ev = emb[(size_t)c * DDIM + d];
        const long a = base + (long)d * 1024 + r;
        const float zv = z[a];
        zq_out[a] = ev;                                   // coalesced 64B runs
        const float df = zv - ev;
        lsum += df * df;
        atomicAdd(&embed_sum[(size_t)c * DDIM + d], zv);  // segment_sum(flat, idx)
    }
    sred[t] = lsum;
    __syncthreads();
    for (int s = 128; s > 0; s >>= 1) {
        if (t < s) sred[t] += sred[t + s];
        __syncthreads();
    }
    if (t == 0) atomicAdd(loss_acc, sred[0]);
}

// -------------------------------------------------------------------------
// Kernel 4: new_cs, n = sum(new_cs), finalize loss.
__global__ __launch_bounds__(256) void vq_ema_cs(const float* __restrict__ ema_cs,
                                                 const float* __restrict__ cluster,
                                                 float* __restrict__ out_newcs,
                                                 float* __restrict__ n_acc,
                                                 const float* __restrict__ loss_acc,
                                                 float* __restrict__ out_loss) {
    __shared__ float sred[256];
    const int t = threadIdx.x;
    const int k = blockIdx.x * 256 + t;      // grid 32 -> 8192
    const float cs = DECAYF * ema_cs[k] + OMDEC * cluster[k];
    out_newcs[k] = cs;
    sred[t] = cs;
    __syncthreads();
    for (int s = 128; s > 0; s >>= 1) {
        if (t < s) sred[t] += sred[t + s];
        __syncthreads();
    }
    if (t == 0) atomicAdd(n_acc, sred[0]);
    if (blockIdx.x == 0 && t == 0)
        *out_loss = BETAF * (*loss_acc) / 2097152.0f;   // mean over N*D
}

// -------------------------------------------------------------------------
// Kernel 5: new_w and normalized new_embedding.
__global__ __launch_bounds__(256) void vq_ema_w(const float* __restrict__ ema_w,
                                                const float* __restrict__ embed_sum,
                                                const float* __restrict__ newcs,
                                                const float* __restrict__ n_acc,
                                                float* __restrict__ out_neww,
                                                float* __restrict__ out_newemb) {
    const size_t e = (size_t)blockIdx.x * 256 + threadIdx.x;  // grid 4096 -> K*D
    const int k = (int)(e >> 7);
    const float nw = DECAYF * ema_w[e] + OMDEC * embed_sum[e];
    out_neww[e] = nw;
    const float n = *n_acc;
    const float denom = (newcs[k] + EPSF) / (n + (float)KCODES * EPSF) * n;
    out_newemb[e] = nw / denom;
}

// -------------------------------------------------------------------------
extern "C" void kernel_launch(void* const* d_in, const int* in_sizes, int n_in,
                              void* d_out, int out_size, void* d_ws, size_t ws_size,
                              hipStream_t stream) {
    const float* z      = (const float*)d_in[0];  // [16,128,32,32]
    const float* emb    = (const float*)d_in[1];  // [8192,128]
    const float* ema_cs = (const float*)d_in[2];  // [8192]
    const float* ema_w  = (const float*)d_in[3];  // [8192,128]

    float* out        = (float*)d_out;
    float* zq_out     = out + OUT_ZQ;
    int*   out_idx    = (int*)(out + OUT_IDX);
    float* out_loss   = out + OUT_LOSS;
    float* out_newemb = out + OUT_NEWEMB;
    float* out_newcs  = out + OUT_NEWCS;
    float* out_neww   = out + OUT_NEWW;

    float* ws         = (float*)d_ws;
    float* ws_cluster = ws + WS_CLUSTER;
    float* ws_embsum  = ws + WS_EMBSUM;
    float* ws_loss    = ws + WS_LOSS;
    float* ws_n       = ws + WS_N;
    float* ws_w2      = ws + WS_W2;

    vq_zero<<<2048, 256, 0, stream>>>(ws, (unsigned long long)WS_ZERO_CT);
    vq_w2<<<1024, 256, 0, stream>>>(emb, ws_w2);
    vq_dist_argmin<<<1024, 256, 0, stream>>>(z, emb, ws_w2, out_idx);
    vq_gather_scatter<<<1024, 256, 0, stream>>>(z, emb, out_idx, zq_out,
                                                ws_embsum, ws_cluster, ws_loss);
    vq_ema_cs<<<32, 256, 0, stream>>>(ema_cs, ws_cluster, out_newcs, ws_n,
                                      ws_loss, out_loss);
    vq_ema_w<<<4096, 256, 0, stream>>>(ema_w, ws_embsum, out_newcs, ws_n,
                                       out_neww, out_newemb);
}